// OverlapsNMS_35416300323286
// MI455X (gfx1250) — compile-verified
//
#include <hip/hip_runtime.h>
#include <cstdint>

// Problem constants (match reference)
#define BATCH   8
#define NBOX    3000
#define NSORT   4096          // bitonic pad
#define NPADC   3072          // padded sorted-column count (96 words * 32)
#define NWORDS  96            // alive/mask words per row
#define MAXOUT  300
#define OV_THR  0.5f
#define SC_THR  0.3f

// ws layout (bytes)
#define WS_M_OFF     0                         // 8 ints
#define WS_ORDER_OFF 256                       // 8*3072 ints
#define WS_MASK_OFF  (256 + BATCH*NPADC*4)     // 8*3000*96 u32

// out layout (floats): sel[8][300][4] | mask[8][300] | idx[8][300]
#define OUT_SEL  0
#define OUT_MSK  (BATCH*MAXOUT*4)
#define OUT_IDX  (BATCH*MAXOUT*4 + BATCH*MAXOUT)

#if defined(__HIP_DEVICE_COMPILE__) && __has_builtin(__builtin_amdgcn_tensor_load_to_lds) && __has_builtin(__builtin_amdgcn_s_wait_tensorcnt)
#define USE_TDM 1
#else
#define USE_TDM 0
#endif

typedef unsigned int v4u __attribute__((ext_vector_type(4)));
typedef int          v8i __attribute__((ext_vector_type(8)));
typedef int          v4i __attribute__((ext_vector_type(4)));

__device__ inline unsigned ballot32(bool p) {
#if defined(__HIP_DEVICE_COMPILE__) && __has_builtin(__builtin_amdgcn_ballot_w32)
  return __builtin_amdgcn_ballot_w32(p);
#else
  return (unsigned)__ballot(p);
#endif
}

// ---------------------------------------------------------------------------
// Phase A: per-batch descending stable sort of scores (score>SC_THR only).
// Inverted-key trick: kinv = (~fbits << 32) | orig_idx ; ascending bitonic sort
// => descending score, ties broken by smaller original index (stable argsort).
// Dead/pad entries get kinv = ~0 and sink to the end.
// ---------------------------------------------------------------------------
__global__ __launch_bounds__(1024)
void nms_sort_kernel(const float* __restrict__ scores,
                     int* __restrict__ Mout, int* __restrict__ order) {
  const int b = blockIdx.x;
  const int t = threadIdx.x;
  __shared__ unsigned long long key[NSORT];
  __shared__ int cnt;
  if (t == 0) cnt = 0;
  __syncthreads();

  int myAlive = 0;
  for (int i = t; i < NSORT; i += 1024) {
    unsigned long long k = ~0ull;
    if (i < NBOX) {
      float s = scores[b * NBOX + i];
      if (s > SC_THR) {
        unsigned fb = __float_as_uint(s);   // s>0 -> monotonic as uint
        k = (((unsigned long long)(~fb)) << 32) | (unsigned)i;
        myAlive++;
      }
    }
    key[i] = k;
  }
  if (myAlive) atomicAdd(&cnt, myAlive);
  __syncthreads();

  // bitonic ascending sort of kinv
  for (unsigned kk = 2; kk <= NSORT; kk <<= 1) {
    for (unsigned j = kk >> 1; j > 0; j >>= 1) {
      for (unsigned i = t; i < NSORT; i += 1024) {
        unsigned ixj = i ^ j;
        if (ixj > i) {
          unsigned long long a = key[i], c = key[ixj];
          bool asc = ((i & kk) == 0);
          if (asc ? (a > c) : (a < c)) { key[i] = c; key[ixj] = a; }
        }
      }
      __syncthreads();
    }
  }

  for (int i = t; i < NPADC; i += 1024) {
    unsigned long long k = key[i];
    int o = ((unsigned)(k >> 32) == 0xFFFFFFFFu) ? 0 : (int)(unsigned)k;
    order[b * NPADC + i] = o;
  }
  if (t == 0) Mout[b] = cnt;
}

// ---------------------------------------------------------------------------
// Phase B: suppression bitmask. One workgroup per (batch, sorted row i).
// Stage the 12KB overlap row into LDS (TDM async DMA when available), gather
// it in sorted-column order, ballot 32 columns/wave into mask words.
// ---------------------------------------------------------------------------
__global__ __launch_bounds__(256)
void nms_mask_kernel(const float* __restrict__ overlaps,
                     const int* __restrict__ Mws,
                     const int* __restrict__ order,
                     unsigned* __restrict__ maskws) {
  const int b  = blockIdx.y;
  const int i  = blockIdx.x;
  const int Mb = min(Mws[b], NBOX);
  if (i >= Mb) return;

  __shared__ float rowbuf[NBOX];
  const int t    = threadIdx.x;
  const int lane = t & 31;
  const int wave = t >> 5;

  const int r_orig = order[b * NPADC + i];
  const float* src = overlaps + ((size_t)b * NBOX + (size_t)r_orig) * NBOX;

#if USE_TDM
  if (t < 32) {  // one wave issues the DMA (TENSORcnt tracked per-wave)
    unsigned long long ga = (unsigned long long)(uintptr_t)src;
    unsigned ldsoff = (unsigned)(uintptr_t)&rowbuf[0];  // low 32 bits = LDS offset
    v4u g0; v8i g1; v4i g2 = {0,0,0,0}, g3 = {0,0,0,0};
    v8i g4 = {0,0,0,0,0,0,0,0};
    g0[0] = 1u;                                   // count=1, user descriptor
    g0[1] = ldsoff;                               // lds_addr
    g0[2] = (unsigned)ga;                         // global_addr[31:0]
    g0[3] = (unsigned)(ga >> 32) | (2u << 30);    // global_addr[56:32] | type=2
    g1[0] = (int)(2u << 16);                      // data_size=4B, mask=0
    g1[1] = (int)(((unsigned)NBOX & 0xFFFFu) << 16);  // tensor_dim0 lo16
    g1[2] = (int)(((unsigned)NBOX >> 16) | (1u << 16)); // tensor_dim0 hi16 | tensor_dim1 lo16 (=1)
    g1[3] = (int)(((unsigned)NBOX & 0xFFFFu) << 16);  // tensor_dim1 hi (=0) | tile_dim0=3000
    g1[4] = 1;                                    // tile_dim1=1, tile_dim2=0
    g1[5] = NBOX;                                 // tensor_dim0_stride lo32
    g1[6] = (int)(((unsigned)NBOX & 0xFFFFu) << 16);  // dim0_stride hi16(=0) | dim1_stride lo16
    g1[7] = 0;
    __builtin_amdgcn_tensor_load_to_lds(g0, g1, g2, g3, g4, 0);
    __builtin_amdgcn_s_wait_tensorcnt(0);
  }
#else
  for (int c = t; c < NBOX; c += 256) rowbuf[c] = src[c];
#endif
  __syncthreads();

  unsigned* maskrow = maskws + ((size_t)b * NBOX + (size_t)i) * NWORDS;
  for (int base = 0; base < NPADC; base += 256) {
    int j = base + t;
    float v = 0.f;
    if (j < Mb) {
      int co = order[b * NPADC + j];
      v = rowbuf[co];
    }
    bool suppress = (j > i) && (v > OV_THR);
    unsigned bal = ballot32(suppress);
    if (lane == 0) maskrow[(base >> 5) + wave] = bal;
  }
}

// ---------------------------------------------------------------------------
// Phase C: serial greedy bit-scan. One wave per batch; 96 alive words held as
// 3 regs/lane. Each step: wave-min over ffs -> selected sorted pos; AND with
// its precomputed mask row; prefetch likely next rows to hide L2 latency.
// ---------------------------------------------------------------------------
__global__ __launch_bounds__(32)
void nms_select_kernel(const int* __restrict__ Mws,
                       const int* __restrict__ order,
                       const unsigned* __restrict__ maskws,
                       const float* __restrict__ attrs,
                       float* __restrict__ out) {
  const int b    = blockIdx.x;
  const int lane = threadIdx.x;
  const int Mb   = min(Mws[b], NBOX);

  // zero-fill this batch's output region (harness poisons d_out)
  for (int x = lane; x < MAXOUT * 4; x += 32) out[OUT_SEL + b * MAXOUT * 4 + x] = 0.f;
  for (int x = lane; x < MAXOUT; x += 32) {
    out[OUT_MSK + b * MAXOUT + x] = 0.f;
    out[OUT_IDX + b * MAXOUT + x] = 0.f;
  }

  // alive words: word w covers sorted positions [w*32, w*32+32); lane owns w = lane + 32*k
  unsigned aw[3];
#pragma unroll
  for (int k = 0; k < 3; ++k) {
    int w = lane + 32 * k;
    int lo = w * 32;
    unsigned v;
    if (lo + 32 <= Mb)      v = 0xFFFFFFFFu;
    else if (lo >= Mb)      v = 0u;
    else                    v = (1u << (Mb - lo)) - 1u;
    aw[k] = v;
  }

  for (int m = 0; m < MAXOUT; ++m) {
    int best = 0x7FFFFFFF;
#pragma unroll
    for (int k = 0; k < 3; ++k)
      if (aw[k]) best = min(best, (lane + 32 * k) * 32 + __builtin_ctz(aw[k]));
#pragma unroll
    for (int off = 16; off > 0; off >>= 1)
      best = min(best, __shfl_xor(best, off, 32));
    if (best >= NPADC) break;   // no survivors -> rest stays zero

    const int i    = best;
    const int orig = order[b * NPADC + i];
    if (lane < 4)
      out[OUT_SEL + (b * MAXOUT + m) * 4 + lane] = attrs[((size_t)b * NBOX + orig) * 4 + lane];
    if (lane == 0) {
      out[OUT_MSK + b * MAXOUT + m] = 1.0f;
      out[OUT_IDX + b * MAXOUT + m] = (float)orig;
    }

    // suppress via precomputed mask row
    const unsigned* row = maskws + ((size_t)b * NBOX + (size_t)i) * NWORDS;
#pragma unroll
    for (int k = 0; k < 3; ++k) aw[k] &= ~row[lane + 32 * k];
    // clear selected bit itself
    {
      int wi = i >> 5;
      if ((wi & 31) == lane) aw[wi >> 5] &= ~(1u << (i & 31));
    }

    // prefetch each lane's own next-candidate mask row (plausible next picks)
    int nb = 0x7FFFFFFF;
#pragma unroll
    for (int k = 0; k < 3; ++k)
      if (aw[k]) nb = min(nb, (lane + 32 * k) * 32 + __builtin_ctz(aw[k]));
    if (nb < Mb)
      __builtin_prefetch(maskws + ((size_t)b * NBOX + (size_t)nb) * NWORDS, 0, 1);
  }
}

// ---------------------------------------------------------------------------
extern "C" void kernel_launch(void* const* d_in, const int* in_sizes, int n_in,
                              void* d_out, int out_size, void* d_ws, size_t ws_size,
                              hipStream_t stream) {
  (void)in_sizes; (void)n_in; (void)out_size; (void)ws_size;
  const float* overlaps = (const float*)d_in[0];
  const float* scores   = (const float*)d_in[1];
  const float* attrs    = (const float*)d_in[2];
  float* out = (float*)d_out;

  int*      Mws     = (int*)((char*)d_ws + WS_M_OFF);
  int*      orderws = (int*)((char*)d_ws + WS_ORDER_OFF);
  unsigned* maskws  = (unsigned*)((char*)d_ws + WS_MASK_OFF);

  nms_sort_kernel<<<BATCH, 1024, 0, stream>>>(scores, Mws, orderws);
  nms_mask_kernel<<<dim3(NBOX, BATCH), 256, 0, stream>>>(overlaps, Mws, orderws, maskws);
  nms_select_kernel<<<BATCH, 32, 0, stream>>>(Mws, orderws, maskws, attrs, out);
}